// CRFAttentionLayer_80152679678280
// MI455X (gfx1250) — compile-verified
//
#include <hip/hip_runtime.h>
#include <math.h>

// CRF attention layer, fused flash-style two-pass softmax, fp32 WMMA (16x16x4).
// N=8192, D=256, K=2, ALPHA=50, BETA=1.  sum(softmax)=1 => denom == 51 exactly.
// Workspace layout (floats), total ~10.15M floats (~41 MB):
//   HpT   : D*N      k-major projection, k-pairs interleaved (coalesced frags)
//   mpart : 8*N      per-chunk row max partials
//   lpart : 8*N      per-chunk row sumexp partials
//   mrow  : N        merged row max
//   lrow  : N        merged row sumexp
//   Opart : 4*N*D    per-chunk P@H partials

typedef __attribute__((ext_vector_type(2))) float v2f;
typedef __attribute__((ext_vector_type(8))) float v8f;

constexpr int NN = 8192;
constexpr int DD = 256;
constexpr int CS1 = 8;   // column chunks for stats pass
constexpr int CS2 = 4;   // column chunks for attention pass

__device__ __forceinline__ v8f wmma4(v2f a, v2f b, v8f c) {
  // D(16x16,f32) = A(16x4,f32) x B(4x16,f32) + C
  return __builtin_amdgcn_wmma_f32_16x16x4_f32(false, a, false, b, (short)0, c,
                                               false, false);
}

// HpT pair-interleaved index: element (k, c) of Hp-transposed (k = feature, c = row)
__device__ __forceinline__ size_t hpT_idx(int k, int c) {
  return (size_t)(k >> 1) * (2 * NN) + (size_t)c * 2 + (k & 1);
}

// ---------------------------------------------------------------------------
// Hp = Hin @ W^T + b, stored transposed (k-major, pair-interleaved) in HpT.
// One wave per 16x16 output tile.  (N/16)*(D/16) = 8192 waves.
__global__ void proj_kernel(const float* __restrict__ Hin,
                            const float* __restrict__ W,
                            const float* __restrict__ bias,
                            float* __restrict__ HpT) {
  int wave = blockIdx.x * (blockDim.x >> 5) + (threadIdx.x >> 5);
  int lane = threadIdx.x & 31;
  int half = lane >> 4, l16 = lane & 15;
  int ib = (wave >> 4) << 4;   // row block (over N)
  int jb = (wave & 15) << 4;   // col block (over D)

  v8f acc = {};
  for (int kk = 0; kk < DD; kk += 4) {
    int k0 = kk + half * 2;
    v2f a, b;
    a.x = Hin[(size_t)(ib + l16) * DD + k0];
    a.y = Hin[(size_t)(ib + l16) * DD + k0 + 1];
    b.x = W[(size_t)(jb + l16) * DD + k0];     // B[k][n] = W[jb+n][k]
    b.y = W[(size_t)(jb + l16) * DD + k0 + 1];
    acc = wmma4(a, b, acc);
  }
  float bv = bias[jb + l16];
#pragma unroll
  for (int r = 0; r < 8; ++r) {
    int i = ib + r + half * 8;   // row of Hp
    int j = jb + l16;            // feature
    HpT[hpT_idx(j, i)] = acc[r] + bv;
  }
}

// ---------------------------------------------------------------------------
// Pass 1: per-row (max, sumexp) over a column chunk, online scalar merge.
// One wave per (16-row block, chunk).  512*CS1 = 4096 waves.
__global__ void stats_partial(const float* __restrict__ HpT,
                              const float* __restrict__ sim,
                              float* __restrict__ mpart,
                              float* __restrict__ lpart) {
  int wave = blockIdx.x * (blockDim.x >> 5) + (threadIdx.x >> 5);
  int lane = threadIdx.x & 31;
  int half = lane >> 4, l16 = lane & 15;
  int rb = (wave >> 3) << 4;
  int ch = wave & 7;
  int cbeg = ch * (NN / CS1);

  // cache the wave's A fragments (16 rows x 256 feats) in 128 VGPRs
  v2f afrag[DD / 4];
#pragma unroll
  for (int t = 0; t < DD / 4; ++t) {
    int k0 = 4 * t + half * 2;
    const float* p = &HpT[hpT_idx(k0, rb + l16)];
    afrag[t].x = p[0];
    afrag[t].y = p[1];
  }

  float m[8], l[8];
#pragma unroll
  for (int r = 0; r < 8; ++r) { m[r] = -3.0e38f; l[r] = 0.0f; }

  for (int ct = 0; ct < (NN / CS1) / 16; ++ct) {
    int c0 = cbeg + ct * 16;
    v8f s = {};
#pragma unroll
    for (int t = 0; t < DD / 4; ++t) {
      int k0 = 4 * t + half * 2;
      const float* p = &HpT[hpT_idx(k0, c0 + l16)];
      v2f bf; bf.x = p[0]; bf.y = p[1];
      s = wmma4(afrag[t], bf, s);
    }
#pragma unroll
    for (int r = 0; r < 8; ++r) {
      int row = rb + r + half * 8;
      int col = c0 + l16;
      float sv = __builtin_nontemporal_load(&sim[(size_t)row * NN + col]);
      float w = (sv != 0.0f) ? 1.0f : 0.0f;
      float se = (w != 0.0f) ? s[r] : -3.0e38f;
      float mn = fmaxf(m[r], se);
      l[r] = l[r] * __expf(m[r] - mn) + w * __expf(se - mn);
      m[r] = mn;
    }
  }
  // merge across the 16 lanes of each half (each half owns 8 rows)
#pragma unroll
  for (int r = 0; r < 8; ++r) {
    for (int off = 1; off < 16; off <<= 1) {
      float m2 = __shfl_xor(m[r], off, 16);
      float l2 = __shfl_xor(l[r], off, 16);
      float mn = fmaxf(m[r], m2);
      l[r] = l[r] * __expf(m[r] - mn) + l2 * __expf(m2 - mn);
      m[r] = mn;
    }
  }
  if (l16 == 0) {
#pragma unroll
    for (int r = 0; r < 8; ++r) {
      int row = rb + r + half * 8;
      mpart[(size_t)ch * NN + row] = m[r];
      lpart[(size_t)ch * NN + row] = l[r];
    }
  }
}

// merge chunk partials -> per-row (m, l)
__global__ void stats_merge(const float* __restrict__ mpart,
                            const float* __restrict__ lpart,
                            float* __restrict__ mrow,
                            float* __restrict__ lrow) {
  int i = blockIdx.x * blockDim.x + threadIdx.x;
  if (i >= NN) return;
  float m = -3.0e38f;
#pragma unroll
  for (int ch = 0; ch < CS1; ++ch) m = fmaxf(m, mpart[(size_t)ch * NN + i]);
  float l = 0.0f;
#pragma unroll
  for (int ch = 0; ch < CS1; ++ch)
    l += lpart[(size_t)ch * NN + i] * __expf(mpart[(size_t)ch * NN + i] - m);
  mrow[i] = m;
  lrow[i] = l;
}

// ---------------------------------------------------------------------------
// Pass 2: O_chunk = sum over chunk columns of exp(S - m) * V  (V = Hin rows).
// One wave per (16-row block, chunk).  512*CS2 = 2048 waves, 128 regs of O.
__global__ void attn_partial(const float* __restrict__ HpT,
                             const float* __restrict__ sim,
                             const float* __restrict__ Hin,
                             const float* __restrict__ mrowg,
                             float* __restrict__ Opart) {
  __shared__ float pT[4][16 * 16];   // wave-private P-tile transpose buffers
  int wv = threadIdx.x >> 5;
  int wave = blockIdx.x * (blockDim.x >> 5) + wv;
  int lane = threadIdx.x & 31;
  int half = lane >> 4, l16 = lane & 15;
  int rb = (wave >> 2) << 4;
  int ch = wave & 3;
  int cbeg = ch * (NN / CS2);

  float mrow[8];
#pragma unroll
  for (int r = 0; r < 8; ++r) mrow[r] = mrowg[rb + r + half * 8];

  v8f o[16];
#pragma unroll
  for (int nt = 0; nt < 16; ++nt) o[nt] = (v8f){};

  for (int ct = 0; ct < (NN / CS2) / 16; ++ct) {
    int c0 = cbeg + ct * 16;
    // scores for this 16x16 tile
    v8f s = {};
#pragma unroll 8
    for (int t = 0; t < DD / 4; ++t) {
      int k0 = 4 * t + half * 2;
      const float* pa = &HpT[hpT_idx(k0, rb + l16)];
      const float* pb = &HpT[hpT_idx(k0, c0 + l16)];
      v2f af; af.x = pa[0]; af.y = pa[1];
      v2f bf; bf.x = pb[0]; bf.y = pb[1];
      s = wmma4(af, bf, s);
    }
    // P = mask * exp(s - m), stored transposed to LDS (kp-major) as A operand
#pragma unroll
    for (int r = 0; r < 8; ++r) {
      int row = rb + r + half * 8;
      int col = c0 + l16;
      float sv = __builtin_nontemporal_load(&sim[(size_t)row * NN + col]);
      float p = (sv != 0.0f) ? __expf(s[r] - mrow[r]) : 0.0f;
      pT[wv][l16 * 16 + r + half * 8] = p;
    }
    asm volatile("s_wait_dscnt 0" ::: "memory");
    // O += P(16x16) @ V(16x256)
#pragma unroll
    for (int t = 0; t < 4; ++t) {
      int kpp = 4 * t + half * 2;
      v2f af;
      af.x = pT[wv][kpp * 16 + l16];
      af.y = pT[wv][(kpp + 1) * 16 + l16];
#pragma unroll
      for (int nt = 0; nt < 16; ++nt) {
        v2f bf;
        bf.x = Hin[(size_t)(c0 + kpp) * DD + nt * 16 + l16];
        bf.y = Hin[(size_t)(c0 + kpp + 1) * DD + nt * 16 + l16];
        o[nt] = wmma4(af, bf, o[nt]);
      }
    }
    asm volatile("s_wait_dscnt 0" ::: "memory");
  }
#pragma unroll
  for (int nt = 0; nt < 16; ++nt) {
#pragma unroll
    for (int r = 0; r < 8; ++r) {
      int row = rb + r + half * 8;
      int feat = nt * 16 + l16;
      Opart[((size_t)ch * NN + row) * DD + feat] = o[nt][r];
    }
  }
}

// H_next = (ALPHA*Q + (sum_ch Opart)/l) / (ALPHA + BETA)
__global__ void finalize_kernel(const float* __restrict__ Opart,
                                const float* __restrict__ lrow,
                                const float* __restrict__ Q,
                                float* __restrict__ Hout) {
  size_t idx = (size_t)blockIdx.x * blockDim.x + threadIdx.x;
  if (idx >= (size_t)NN * DD) return;
  size_t row = idx / DD;
  float acc = 0.0f;
#pragma unroll
  for (int ch = 0; ch < CS2; ++ch) acc += Opart[(size_t)ch * NN * DD + idx];
  Hout[idx] = (50.0f * Q[idx] + acc / lrow[row]) * (1.0f / 51.0f);
}

extern "C" void kernel_launch(void* const* d_in, const int* in_sizes, int n_in,
                              void* d_out, int out_size, void* d_ws, size_t ws_size,
                              hipStream_t stream) {
  const float* Q   = (const float*)d_in[0];
  const float* sim = (const float*)d_in[1];
  const float* W   = (const float*)d_in[2];
  const float* b   = (const float*)d_in[3];
  float* out = (float*)d_out;
  float* ws  = (float*)d_ws;

  size_t off = 0;
  float* HpT   = ws + off; off += (size_t)DD * NN;        // 2M floats
  float* mpart = ws + off; off += (size_t)CS1 * NN;
  float* lpart = ws + off; off += (size_t)CS1 * NN;
  float* mrow  = ws + off; off += NN;
  float* lrow  = ws + off; off += NN;
  float* Opart = ws + off; off += (size_t)CS2 * NN * DD;  // 8M floats

  for (int it = 0; it < 2; ++it) {
    const float* Hin = (it == 0) ? Q : out;  // d_out doubles as H1 buffer
    float* Hout = out;

    // Hp^T (pair-interleaved)
    proj_kernel<<<(NN / 16) * (DD / 16) / 8, 256, 0, stream>>>(Hin, W, b, HpT);
    // row max / sumexp
    stats_partial<<<(NN / 16) * CS1 / 8, 256, 0, stream>>>(HpT, sim, mpart, lpart);
    stats_merge<<<NN / 256, 256, 0, stream>>>(mpart, lpart, mrow, lrow);
    // P @ H partials
    attn_partial<<<(NN / 16) * CS2 / 4, 128, 0, stream>>>(HpT, sim, Hin, mrow, Opart);
    // combine, normalize
    finalize_kernel<<<(NN * DD) / 256, 256, 0, stream>>>(Opart, lrow, Q, Hout);
  }
}